// MSELoss_40767829573845
// MI455X (gfx1250) — compile-verified
//
#include <hip/hip_runtime.h>
#include <hip/hip_bf16.h>
#include <stdint.h>

// Problem constants (match reference)
#define H_ 512
#define W_ 512
#define HW (H_ * W_)
#define BATCH 16
#define TOPK 512
#define CAP 16384          // candidate cap per image (expected ~10.5K NMS survivors)

typedef float v2f __attribute__((ext_vector_type(2)));
typedef float v8f __attribute__((ext_vector_type(8)));

#if __has_builtin(__builtin_amdgcn_wmma_f32_16x16x4_f32)
#define HAVE_WMMA_F32X4 1
#else
#define HAVE_WMMA_F32X4 0
#endif

// Normalized 7-tap Gaussian (sigma=2): weight for |d| in [0,3], 0 outside.
// Branchless select chain -> pure VALU (v_cmp + v_cndmask), no table loads.
__device__ __forceinline__ float gw(int d) {
    int ad = d < 0 ? -d : d;
    float w = 0.0f;
    w = (ad == 3) ? 0.0701593f : w;
    w = (ad == 2) ? 0.1310749f : w;
    w = (ad == 1) ? 0.1907131f : w;
    w = (ad == 0) ? 0.2161060f : w;
    return w;
}

// ---------------------------------------------------------------- init
__global__ void k_zero_f32(float* __restrict__ p, int n) {
    int i = blockIdx.x * blockDim.x + threadIdx.x;
    if (i < n) p[i] = 0.0f;
}

__global__ void k_init_small(int* __restrict__ counters, double* __restrict__ accum) {
    int i = threadIdx.x;
    if (i < 32) counters[i] = 0;
    if (i < 2) accum[i] = 0.0;
}

// ---------------------------------------------------------------- warp (bilinear homography sample + vis mask)
__global__ void k_warp(const float* __restrict__ s2, const float* __restrict__ homo,
                       float* __restrict__ w2, unsigned char* __restrict__ vis) {
    int i = blockIdx.x * 256 + threadIdx.x;          // exactly BATCH*HW threads
    int b = i >> 18;
    int rem = i & (HW - 1);
    int y = rem >> 9, x = rem & (W_ - 1);
    const float* Hm = homo + b * 9;
    float fx = (float)x, fy = (float)y;
    float px = Hm[0] * fx + Hm[1] * fy + Hm[2];
    float py = Hm[3] * fx + Hm[4] * fy + Hm[5];
    float pz = Hm[6] * fx + Hm[7] * fy + Hm[8] + 1e-8f;
    float sx = px / pz, sy = py / pz;
    float x0 = floorf(sx), y0 = floorf(sy);
    float wx1 = sx - x0, wx0 = 1.0f - wx1;
    float wy1 = sy - y0, wy0 = 1.0f - wy1;
    const float* img = s2 + (size_t)b * HW;
    float acc = 0.0f, msk = 0.0f;
#pragma unroll
    for (int cy = 0; cy < 2; ++cy) {
#pragma unroll
        for (int cx = 0; cx < 2; ++cx) {
            float xi = x0 + (float)cx, yi = y0 + (float)cy;
            float wgt = (cx ? wx1 : wx0) * (cy ? wy1 : wy0);
            bool v = (xi >= 0.0f) && (xi <= (float)(W_ - 1)) && (yi >= 0.0f) && (yi <= (float)(H_ - 1));
            int xc = (int)fminf(fmaxf(xi, 0.0f), (float)(W_ - 1));
            int yc = (int)fminf(fmaxf(yi, 0.0f), (float)(H_ - 1));
            float s = v ? img[yc * W_ + xc] : 0.0f;
            acc += wgt * s;
            msk += wgt * (v ? 1.0f : 0.0f);
        }
    }
    w2[i] = acc;
    vis[i] = (msk > 0.0f) ? 1 : 0;
}

// ---------------------------------------------------------------- 5x5 NMS + candidate compaction
__global__ void k_nms_collect(const float* __restrict__ src, float* __restrict__ cval,
                              int* __restrict__ cidx, int* __restrict__ counter) {
    __shared__ float t[20][20];
    int b = blockIdx.z;
    int x0 = blockIdx.x * 16, y0 = blockIdx.y * 16;
    const float* img = src + (size_t)b * HW;
    int tid = threadIdx.y * 16 + threadIdx.x;
    for (int i = tid; i < 400; i += 256) {
        int ly = i / 20, lx = i % 20;
        int gy = y0 + ly - 2, gx = x0 + lx - 2;
        t[ly][lx] = (gx >= 0 && gx < W_ && gy >= 0 && gy < H_) ? img[gy * W_ + gx] : -3.402823466e38f;
    }
    __syncthreads();
    float v = t[threadIdx.y + 2][threadIdx.x + 2];
    float mx = v;
#pragma unroll
    for (int dy = 0; dy < 5; ++dy)
#pragma unroll
        for (int dx = 0; dx < 5; ++dx) mx = fmaxf(mx, t[threadIdx.y + dy][threadIdx.x + dx]);
    if (v == mx && v > 0.0f) {   // NMS_THRESH == 0, strict >
        int slot = atomicAdd(counter + b, 1);
        if (slot < CAP) {
            cval[b * CAP + slot] = v;
            cidx[b * CAP + slot] = (y0 + (int)threadIdx.y) * W_ + (x0 + (int)threadIdx.x);
        }
    }
}

// ---------------------------------------------------------------- exact top-512 per image
// Binary search the 512th value over IEEE bit patterns (candidates are all > 0,
// so uint-bit order == float order). Ties resolved by lowest flat index, like
// jax.lax.top_k. Result bitonic-sorted descending on packed u64 keys.
__global__ void k_select(const float* __restrict__ cval, const int* __restrict__ cidx,
                         const int* __restrict__ counter, int mode,
                         float* __restrict__ out, float* __restrict__ gt) {
    const int b = blockIdx.x;
    const int tid = threadIdx.x;
    const float* cv = cval + b * CAP;
    const int* ci = cidx + b * CAP;
    __shared__ int s_red, s_cnt, s_eq;
    __shared__ unsigned long long keys[TOPK];
    __shared__ int eq[1024];

    int n = counter[b];
    if (n > CAP) n = CAP;

    auto countGE = [&](unsigned t) -> int {
        if (tid == 0) s_red = 0;
        __syncthreads();
        int c = 0;
        for (int i = tid; i < n; i += 256) c += (__float_as_uint(cv[i]) >= t) ? 1 : 0;
        for (int off = 16; off; off >>= 1) c += __shfl_down(c, off, 32);
        if ((tid & 31) == 0) atomicAdd(&s_red, c);
        __syncthreads();
        int r = s_red;
        __syncthreads();
        return r;
    };

    // padding key: val=0, encoded idx -> 0  (row 0, col 0; gt-scatter skips vb==0)
    for (int i = tid; i < TOPK; i += 256) keys[i] = 0xFFFFFFFFull;
    for (int i = tid; i < 1024; i += 256) eq[i] = 0x7FFFFFFF;
    if (tid == 0) { s_cnt = 0; s_eq = 0; }
    __syncthreads();

    if (n <= TOPK) {
        // fewer survivors than K: take them all (reference pads with zeros;
        // statistically unreachable for these inputs)
        for (int i = tid; i < n; i += 256) {
            unsigned vb = __float_as_uint(cv[i]);
            int slot = atomicAdd(&s_cnt, 1);
            keys[slot] = ((unsigned long long)vb << 32) | (unsigned long long)(0xFFFFFFFFu - (unsigned)ci[i]);
        }
        __syncthreads();
    } else {
        unsigned lo = 0u, hi = 0x7F800000u;   // count(>=lo)=n>=K ; count(>=inf)=0<K
        while (hi - lo > 1u) {
            unsigned mid = lo + ((hi - lo) >> 1);
            int c = countGE(mid);
            if (c >= TOPK) lo = mid; else hi = mid;
        }
        unsigned v512 = lo;                    // bits of the K-th largest value
        int mGreater = countGE(v512 + 1u);     // count strictly greater
        for (int i = tid; i < n; i += 256) {
            unsigned vb = __float_as_uint(cv[i]);
            if (vb > v512) {
                int slot = atomicAdd(&s_cnt, 1);
                if (slot < TOPK)
                    keys[slot] = ((unsigned long long)vb << 32) | (unsigned long long)(0xFFFFFFFFu - (unsigned)ci[i]);
            } else if (vb == v512) {
                int p = atomicAdd(&s_eq, 1);
                if (p < 1024) eq[p] = ci[i];
            }
        }
        __syncthreads();
        // sort tie indices ascending (bitonic, N=1024)
        for (int k = 2; k <= 1024; k <<= 1)
            for (int j = k >> 1; j > 0; j >>= 1) {
                for (int i = tid; i < 1024; i += 256) {
                    int p = i ^ j;
                    if (p > i) {
                        bool asc = ((i & k) == 0);
                        int a = eq[i], bb = eq[p];
                        if (asc ? (a > bb) : (a < bb)) { eq[i] = bb; eq[p] = a; }
                    }
                }
                __syncthreads();
            }
        int need = TOPK - mGreater;
        for (int j2 = tid; j2 < need; j2 += 256) {
            int idx = (j2 < 1024) ? eq[j2] : 0;
            keys[mGreater + j2] =
                ((unsigned long long)v512 << 32) | (unsigned long long)(0xFFFFFFFFu - (unsigned)idx);
        }
        __syncthreads();
    }

    // bitonic sort keys descending (N=512): value desc, index asc on ties
    for (int k = 2; k <= TOPK; k <<= 1)
        for (int j = k >> 1; j > 0; j >>= 1) {
            for (int i = tid; i < TOPK; i += 256) {
                int p = i ^ j;
                if (p > i) {
                    bool desc = ((i & k) == 0);
                    unsigned long long a = keys[i], bb = keys[p];
                    if (desc ? (a < bb) : (a > bb)) { keys[i] = bb; keys[p] = a; }
                }
            }
            __syncthreads();
        }

    if (mode == 0) {
        // kp1 output: d_out[1 + ((b*512 + s)*2 + {0,1})] = {row, col} as float
        for (int s = tid; s < TOPK; s += 256) {
            unsigned long long kk = keys[s];
            unsigned idx = 0xFFFFFFFFu - (unsigned)(kk & 0xFFFFFFFFull);
            size_t o = 1 + ((size_t)(b * TOPK + s)) * 2;
            out[o + 0] = (float)(idx >> 9);
            out[o + 1] = (float)(idx & (W_ - 1));
        }
    } else {
        // scatter sparse GT (buffer pre-zeroed)
        for (int s = tid; s < TOPK; s += 256) {
            unsigned long long kk = keys[s];
            unsigned vb = (unsigned)(kk >> 32);
            if (vb) {
                unsigned idx = 0xFFFFFFFFu - (unsigned)(kk & 0xFFFFFFFFull);
                gt[(size_t)b * HW + idx] = __uint_as_float(vb);
            }
        }
    }
}

// ---------------------------------------------------------------- separable Gaussian via f32 WMMA
// Row pass: Y = X * M, banded 512x512 M decomposed into 16x16 tiles; only the
// center tile (4 K-chunks) and one K-chunk of each adjacent tile are nonzero
// (half-width 3 < 16). One wave per 16x16 output tile, 6 v_wmma_f32_16x16x4_f32.
// Fragment layouts per CDNA5 ISA §7.12.2:
//  A 16x4 f32 : lane L -> m=L&15, VGPR pair holds K = 2*(L>>4) + {0,1}
//  B 4x16  f32: lane L -> n=L&15, VGPR pair holds rows K = 2*(L>>4) + {0,1}
//  C/D 16x16  : lane L -> n=L&15, VGPR r holds M = r + 8*(L>>4)
__global__ void k_blur_row(const float* __restrict__ src, float* __restrict__ dst) {
    int gtid = blockIdx.x * 256 + threadIdx.x;
    int wave = gtid >> 5;
    int lane = threadIdx.x & 31;
    int tx = wave & 31, ty = (wave >> 5) & 31, b = wave >> 10;
    const float* img = src + (size_t)b * HW;
    float* outp = dst + (size_t)b * HW;
    int m = lane & 15, khalf = lane >> 4;
    v8f acc = {};
#if HAVE_WMMA_F32X4
#pragma unroll
    for (int o = -1; o <= 1; ++o) {
        int sx = tx + o;
        if (sx < 0 || sx > 31) continue;         // uniform per wave -> scalar branch
        int kcLo = (o == 1) ? 0 : ((o == -1) ? 3 : 0);
        int kcHi = (o == 1) ? 0 : 3;
#pragma unroll
        for (int kc = kcLo; kc <= kcHi; ++kc) {
            const float* ap = img + (size_t)(ty * 16 + m) * W_ + sx * 16 + kc * 4 + 2 * khalf;
            v2f a; a.x = ap[0]; a.y = ap[1];
            int k0 = kc * 4 + 2 * khalf;
            v2f bf; bf.x = gw(m - k0 - 16 * o); bf.y = gw(m - (k0 + 1) - 16 * o);
            acc = __builtin_amdgcn_wmma_f32_16x16x4_f32(false, a, false, bf, (short)0, acc, false, false);
        }
    }
#else
#pragma unroll
    for (int r = 0; r < 8; ++r) {
        int yy = ty * 16 + r + 8 * khalf, xx = tx * 16 + m;
        float s = 0.0f;
        for (int d = -3; d <= 3; ++d) {
            int x2 = xx + d;
            if (x2 >= 0 && x2 < W_) s += gw(d) * img[(size_t)yy * W_ + x2];
        }
        acc[r] = s;
    }
#endif
    float* op = outp + (size_t)(ty * 16 + 8 * khalf) * W_ + tx * 16 + m;
#pragma unroll
    for (int r = 0; r < 8; ++r) op[(size_t)r * W_] = acc[r];
}

// Column pass: Z = M^T * Y ; A = coefficient chunks, B = source tile rows (coalesced loads)
__global__ void k_blur_col(const float* __restrict__ src, float* __restrict__ dst) {
    int gtid = blockIdx.x * 256 + threadIdx.x;
    int wave = gtid >> 5;
    int lane = threadIdx.x & 31;
    int tx = wave & 31, ty = (wave >> 5) & 31, b = wave >> 10;
    const float* img = src + (size_t)b * HW;
    float* outp = dst + (size_t)b * HW;
    int m = lane & 15, khalf = lane >> 4;
    v8f acc = {};
#if HAVE_WMMA_F32X4
#pragma unroll
    for (int o = -1; o <= 1; ++o) {
        int sy = ty + o;
        if (sy < 0 || sy > 31) continue;
        int kcLo = (o == 1) ? 0 : ((o == -1) ? 3 : 0);
        int kcHi = (o == 1) ? 0 : 3;
#pragma unroll
        for (int kc = kcLo; kc <= kcHi; ++kc) {
            int k0 = kc * 4 + 2 * khalf;
            v2f a; a.x = gw(16 * o + k0 - m); a.y = gw(16 * o + k0 + 1 - m);
            const float* bp = img + (size_t)(sy * 16 + k0) * W_ + tx * 16 + m;  // n == m here
            v2f bf; bf.x = bp[0]; bf.y = bp[W_];
            acc = __builtin_amdgcn_wmma_f32_16x16x4_f32(false, a, false, bf, (short)0, acc, false, false);
        }
    }
#else
#pragma unroll
    for (int r = 0; r < 8; ++r) {
        int yy = ty * 16 + r + 8 * khalf, xx = tx * 16 + m;
        float s = 0.0f;
        for (int d = -3; d <= 3; ++d) {
            int y2 = yy + d;
            if (y2 >= 0 && y2 < H_) s += gw(d) * img[(size_t)y2 * W_ + xx];
        }
        acc[r] = s;
    }
#endif
    float* op = outp + (size_t)(ty * 16 + 8 * khalf) * W_ + tx * 16 + m;
#pragma unroll
    for (int r = 0; r < 8; ++r) op[(size_t)r * W_] = acc[r];
}

// ---------------------------------------------------------------- fused loss reduction
__global__ void k_loss(const float* __restrict__ s1, const float* __restrict__ gt,
                       const unsigned char* __restrict__ vis, double* __restrict__ accum) {
    __shared__ double sd[2][8];
    float num = 0.0f, den = 0.0f;
    size_t total = (size_t)BATCH * HW;
    for (size_t i = (size_t)blockIdx.x * 256 + threadIdx.x; i < total; i += (size_t)gridDim.x * 256) {
        float mface = (float)vis[i];
        float d = s1[i] - gt[i];
        num += d * d * mface;
        den += mface;
    }
    for (int off = 16; off; off >>= 1) {
        num += __shfl_down(num, off, 32);
        den += __shfl_down(den, off, 32);
    }
    int wid = threadIdx.x >> 5;
    if ((threadIdx.x & 31) == 0) { sd[0][wid] = (double)num; sd[1][wid] = (double)den; }
    __syncthreads();
    if (threadIdx.x == 0) {
        double n2 = 0.0, d2 = 0.0;
        for (int w = 0; w < 8; ++w) { n2 += sd[0][w]; d2 += sd[1][w]; }
        atomicAdd(&accum[0], n2);
        atomicAdd(&accum[1], d2);
    }
}

__global__ void k_final(const double* __restrict__ accum, float* __restrict__ out) {
    if (blockIdx.x == 0 && threadIdx.x == 0) out[0] = (float)(accum[0] / accum[1]);  // LOSS_LAMBDA = 1
}

// ---------------------------------------------------------------- launch
extern "C" void kernel_launch(void* const* d_in, const int* in_sizes, int n_in,
                              void* d_out, int out_size, void* d_ws, size_t ws_size,
                              hipStream_t stream) {
    (void)in_sizes; (void)n_in; (void)out_size; (void)ws_size;
    const float* score1 = (const float*)d_in[0];
    const float* score2 = (const float*)d_in[1];
    const float* homo   = (const float*)d_in[2];
    float* out = (float*)d_out;      // [0] = loss, [1..16384] = kp1 (B,1,512,2) as float

    char* ws = (char*)d_ws;          // needs ~56 MB + 160 B of scratch
    const size_t MB = (size_t)1 << 20;
    float* w2            = (float*)(ws + 0 * MB);            // warped score2        16 MB
    float* gt            = (float*)(ws + 16 * MB);           // sparse->blurred GT   16 MB
    float* tmp           = (float*)(ws + 32 * MB);           // row-blur intermediate 16 MB
    unsigned char* vis   = (unsigned char*)(ws + 48 * MB);   // visibility bytes      4 MB
    float* cval1         = (float*)(ws + 52 * MB);           // candidates score1     1 MB
    int*   cidx1         = (int*)(ws + 53 * MB);             //                       1 MB
    float* cval2         = (float*)(ws + 54 * MB);           // candidates warped     1 MB
    int*   cidx2         = (int*)(ws + 55 * MB);             //                       1 MB
    int*   counters      = (int*)(ws + 56 * MB);             // [0..15] s1, [16..31] w2
    double* accum        = (double*)(ws + 56 * MB + 128);    // [num, den]

    k_zero_f32<<<16384, 256, 0, stream>>>(gt, BATCH * HW);
    k_init_small<<<1, 64, 0, stream>>>(counters, accum);

    k_warp<<<16384, 256, 0, stream>>>(score2, homo, w2, vis);

    dim3 gN(32, 32, 16), bN(16, 16);
    k_nms_collect<<<gN, bN, 0, stream>>>(score1, cval1, cidx1, counters);
    k_nms_collect<<<gN, bN, 0, stream>>>(w2, cval2, cidx2, counters + 16);

    k_select<<<16, 256, 0, stream>>>(cval1, cidx1, counters, 0, out, gt);       // kp1
    k_select<<<16, 256, 0, stream>>>(cval2, cidx2, counters + 16, 1, out, gt);  // scatter GT

    k_blur_row<<<2048, 256, 0, stream>>>(gt, tmp);   // WMMA
    k_blur_col<<<2048, 256, 0, stream>>>(tmp, gt);   // WMMA

    k_loss<<<2048, 256, 0, stream>>>(score1, gt, vis, accum);
    k_final<<<1, 1, 0, stream>>>(accum, out);
}